// HeteroGNN_69801808494760
// MI455X (gfx1250) — compile-verified
//
#include <hip/hip_runtime.h>

typedef __attribute__((ext_vector_type(2))) float v2f;
typedef __attribute__((ext_vector_type(8))) float v8f;

#define DF 128  // feature width for every tensor in this model

// ---------------- utility: zero a float region ----------------
__global__ void __launch_bounds__(256) zero_f32_kernel(float* __restrict__ p, long n) {
    long t = (long)blockIdx.x * blockDim.x + threadIdx.x;
    if (t < n) p[t] = 0.0f;
}

// ---------------- edge scatter: one wave32 per edge ----------------
// Each lane handles 4 contiguous floats of the 128-wide feature row.
__global__ void __launch_bounds__(256) scatter_accum_kernel(
    const float* __restrict__ xsrc, const int* __restrict__ src,
    const int* __restrict__ dst, float* __restrict__ acc,
    float* __restrict__ cnt, int E)
{
    long t = (long)blockIdx.x * blockDim.x + threadIdx.x;
    int e = (int)(t >> 5);
    int lane = (int)(t & 31);
    if (e >= E) return;
    int s = src[e];
    int d = dst[e];
    const float4 v = *(const float4*)(xsrc + (long)s * DF + lane * 4);
    float* a = acc + (long)d * DF + lane * 4;
    atomicAdd(a + 0, v.x);
    atomicAdd(a + 1, v.y);
    atomicAdd(a + 2, v.z);
    atomicAdd(a + 3, v.w);
    if (lane == 0) atomicAdd(cnt + d, 1.0f);
}

// ---------------- mean = sum / max(cnt, 1) ----------------
__global__ void __launch_bounds__(256) finalize_mean_kernel(
    float* __restrict__ acc, const float* __restrict__ cnt, long total)
{
    long t = (long)blockIdx.x * blockDim.x + threadIdx.x;
    if (t >= total) return;
    float c = cnt[t >> 7];            // row = t / 128
    acc[t] = acc[t] / fmaxf(c, 1.0f);
}

// ---------------- fused multi-part GEMM via V_WMMA_F32_16X16X4_F32 ----------------
// out[M x 128] = sum_{p < NPARTS} Ap[M x 128] @ Bp[128 x 128]  (+ bias0 (+ bias1)) (ReLU opt)
// Block: 256 threads = 8 waves. Block owns 64 rows (4 row-tiles of 16, one
// accumulator each => B fragments amortized 4x); wave w owns columns [16w, 16w+16).
// Tail blocks clamp row-tiles onto [M-16, M): they recompute bitwise-identical
// values, so EXEC stays all-ones through every WMMA and stores need no guards.
template <int NPARTS>
__global__ void __launch_bounds__(256) wmma_gemm_kernel(
    const float* __restrict__ A0, const float* __restrict__ B0,
    const float* __restrict__ A1, const float* __restrict__ B1,
    const float* __restrict__ A2, const float* __restrict__ B2,
    const float* __restrict__ A3, const float* __restrict__ B3,
    const float* __restrict__ bias0, const float* __restrict__ bias1,
    float* __restrict__ out, int M, int do_relu)
{
    constexpr int MT = 4;                        // row tiles per block
    const int lane    = threadIdx.x & 31;
    const int colTile = threadIdx.x >> 5;        // 0..7

    // A 16x4 f32 fragment layout: lanes 0-15 carry M=0..15 with K={0,1};
    // lanes 16-31 carry the same M rows with K={2,3}.
    const int mrow = lane & 15;
    const int kgrp = (lane >> 4) * 2;            // 0 or 2
    const int ncol = lane & 15;                  // B/C/D column within tile

    int r0[MT];
#pragma unroll
    for (int t = 0; t < MT; ++t) {
        int r = blockIdx.x * (16 * MT) + t * 16;
        r0[t] = (r + 16 <= M) ? r : (M - 16);    // uniform clamp, no divergence
    }

    v8f acc[MT];
#pragma unroll
    for (int t = 0; t < MT; ++t) acc[t] = (v8f){};

    auto do_part = [&](const float* __restrict__ A, const float* __restrict__ B) {
        const float* bcol = B + colTile * 16 + ncol;
        const float* arow[MT];
#pragma unroll
        for (int t = 0; t < MT; ++t) {
            arow[t] = A + (long)(r0[t] + mrow) * DF;
            __builtin_prefetch(arow[t], 0, 3);   // global_prefetch_b8: warm A rows
        }
#pragma unroll 4
        for (int kk = 0; kk < DF; kk += 4) {
            const int ka = kk + kgrp;
            v2f b;
            b.x = bcol[ka * DF];                 // B[ka][col]
            b.y = bcol[(ka + 1) * DF];           // B[ka+1][col]
#pragma unroll
            for (int t = 0; t < MT; ++t) {
                v2f a = *(const v2f*)(arow[t] + ka);   // A[m][ka..ka+1], 8B aligned
                acc[t] = __builtin_amdgcn_wmma_f32_16x16x4_f32(
                    false, a, false, b, (short)0, acc[t], false, false);
            }
        }
    };

    do_part(A0, B0);
    if (NPARTS >= 2) do_part(A1, B1);
    if (NPARTS >= 4) {
        do_part(A2, B2);
        do_part(A3, B3);
    }

    // C/D layout: VGPR r -> row (r + 8*(lane>>4)), col = lane&15 within tile.
    const int col = colTile * 16 + ncol;
    float bsum = bias0[col];
    if (bias1) bsum += bias1[col];
#pragma unroll
    for (int t = 0; t < MT; ++t) {
        const int rbase = r0[t] + (lane >> 4) * 8;
#pragma unroll
        for (int r = 0; r < 8; ++r) {
            float v = acc[t][r] + bsum;
            if (do_relu) v = fmaxf(v, 0.0f);
            out[(long)(rbase + r) * DF + col] = v;
        }
    }
}

// ---------------- host-side orchestration ----------------
extern "C" void kernel_launch(void* const* d_in, const int* in_sizes, int n_in,
                              void* d_out, int out_size, void* d_ws, size_t ws_size,
                              hipStream_t stream)
{
    const float* x_kw = (const float*)d_in[0];
    const float* x_rt = (const float*)d_in[1];
    const int* src_a = (const int*)d_in[2];
    const int* dst_a = (const int*)d_in[3];
    const int* src_b = (const int*)d_in[4];
    const int* dst_b = (const int*)d_in[5];
    const int* src_c = (const int*)d_in[6];
    const int* dst_c = (const int*)d_in[7];

    // weights: per layer, per relation (a,b,c): Wl, bl, Wr
    const float *Wl1a = (const float*)d_in[8],  *bl1a = (const float*)d_in[9],  *Wr1a = (const float*)d_in[10];
    const float *Wl1b = (const float*)d_in[11], *bl1b = (const float*)d_in[12], *Wr1b = (const float*)d_in[13];
    const float *Wl1c = (const float*)d_in[14], *bl1c = (const float*)d_in[15], *Wr1c = (const float*)d_in[16];
    const float *Wl2a = (const float*)d_in[17], *bl2a = (const float*)d_in[18], *Wr2a = (const float*)d_in[19];
    const float *Wl2b = (const float*)d_in[20], *bl2b = (const float*)d_in[21], *Wr2b = (const float*)d_in[22];
    const float *Wl2c = (const float*)d_in[23], *bl2c = (const float*)d_in[24], *Wr2c = (const float*)d_in[25];

    const int nkw = in_sizes[0] / DF;   // 100000
    const int nrt = in_sizes[1] / DF;   // 20000
    const int Ea = in_sizes[2];
    const int Eb = in_sizes[4];
    const int Ec = in_sizes[6];

    // workspace layout (floats), sums+counts contiguous so one zero pass covers them
    float* ws = (float*)d_ws;
    long off = 0;
    float* sum_a = ws + off; off += (long)nkw * DF;
    float* sum_b = ws + off; off += (long)nkw * DF;
    float* sum_c = ws + off; off += (long)nrt * DF;
    float* cnt_a = ws + off; off += nkw;
    float* cnt_b = ws + off; off += nkw;
    float* cnt_c = ws + off; off += nrt;
    float* kw1   = ws + off; off += (long)nkw * DF;
    float* rt1   = ws + off; off += (long)nrt * DF;

    float* out_kw2 = (float*)d_out;
    float* out_rt2 = out_kw2 + (long)nkw * DF;

    const int TB = 256;
    const long zn = (long)nkw * DF * 2 + (long)nrt * DF + 2L * nkw + nrt;
    auto gz = [](long n) { return (int)((n + 255) / 256); };
    const int gkw = (nkw + 63) / 64;    // 64 rows per GEMM block
    const int grt = (nrt + 63) / 64;

    // ================= layer 1 =================
    zero_f32_kernel<<<gz(zn), TB, 0, stream>>>(sum_a, zn);

    scatter_accum_kernel<<<gz((long)Ea * 32), TB, 0, stream>>>(x_rt, src_a, dst_a, sum_a, cnt_a, Ea);
    scatter_accum_kernel<<<gz((long)Eb * 32), TB, 0, stream>>>(x_kw, src_b, dst_b, sum_b, cnt_b, Eb);
    scatter_accum_kernel<<<gz((long)Ec * 32), TB, 0, stream>>>(x_rt, src_c, dst_c, sum_c, cnt_c, Ec);

    finalize_mean_kernel<<<gz((long)nkw * DF), TB, 0, stream>>>(sum_a, cnt_a, (long)nkw * DF);
    finalize_mean_kernel<<<gz((long)nkw * DF), TB, 0, stream>>>(sum_b, cnt_b, (long)nkw * DF);
    finalize_mean_kernel<<<gz((long)nrt * DF), TB, 0, stream>>>(sum_c, cnt_c, (long)nrt * DF);

    // kw1 = relu(mean_a@Wl1a + bl1a + x_kw@Wr1a + mean_b@Wl1b + bl1b + x_kw@Wr1b)
    wmma_gemm_kernel<4><<<gkw, TB, 0, stream>>>(
        sum_a, Wl1a, x_kw, Wr1a, sum_b, Wl1b, x_kw, Wr1b,
        bl1a, bl1b, kw1, nkw, 1);
    // rt1 = relu(mean_c@Wl1c + bl1c + x_rt@Wr1c)
    wmma_gemm_kernel<2><<<grt, TB, 0, stream>>>(
        sum_c, Wl1c, x_rt, Wr1c, nullptr, nullptr, nullptr, nullptr,
        bl1c, nullptr, rt1, nrt, 1);

    // ================= layer 2 =================
    zero_f32_kernel<<<gz(zn), TB, 0, stream>>>(sum_a, zn);

    scatter_accum_kernel<<<gz((long)Ea * 32), TB, 0, stream>>>(rt1, src_a, dst_a, sum_a, cnt_a, Ea);
    scatter_accum_kernel<<<gz((long)Eb * 32), TB, 0, stream>>>(kw1, src_b, dst_b, sum_b, cnt_b, Eb);
    scatter_accum_kernel<<<gz((long)Ec * 32), TB, 0, stream>>>(rt1, src_c, dst_c, sum_c, cnt_c, Ec);

    finalize_mean_kernel<<<gz((long)nkw * DF), TB, 0, stream>>>(sum_a, cnt_a, (long)nkw * DF);
    finalize_mean_kernel<<<gz((long)nkw * DF), TB, 0, stream>>>(sum_b, cnt_b, (long)nkw * DF);
    finalize_mean_kernel<<<gz((long)nrt * DF), TB, 0, stream>>>(sum_c, cnt_c, (long)nrt * DF);

    // kw2 = mean_a@Wl2a + bl2a + kw1@Wr2a + mean_b@Wl2b + bl2b + kw1@Wr2b   (no relu)
    wmma_gemm_kernel<4><<<gkw, TB, 0, stream>>>(
        sum_a, Wl2a, kw1, Wr2a, sum_b, Wl2b, kw1, Wr2b,
        bl2a, bl2b, out_kw2, nkw, 0);
    // rt2 = mean_c@Wl2c + bl2c + rt1@Wr2c
    wmma_gemm_kernel<2><<<grt, TB, 0, stream>>>(
        sum_c, Wl2c, rt1, Wr2c, nullptr, nullptr, nullptr, nullptr,
        bl2c, nullptr, out_rt2, nrt, 0);
}